// ResidualQuantize_65000035057700
// MI455X (gfx1250) — compile-verified
//
#include <hip/hip_runtime.h>
#include <hip/hip_bf16.h>

// Residual VQ: two fused nearest-neighbor stages over 4096-entry codebooks,
// distances via V_WMMA_F32_16X16X4_F32 (exact f32 accumulate).

typedef __attribute__((ext_vector_type(2))) float v2f;
typedef __attribute__((ext_vector_type(8))) float v8f;

#define RQ_N     131072
#define RQ_C     64
#define RQ_K     4096

// Scan `ncodes` codebook rows against the 16-row A tile held in `a[16]`
// (16x4-f32 A layout per k-step). On exit, lanes with (lane&15)==0 hold the
// per-row best: rows (r + 8*(lane>>4)) in best[r]/bidx[r].
__device__ __forceinline__ void rq_scan(const v2f a[16],
                                        const float* __restrict__ cb,
                                        int ncodes, int lane,
                                        float best[8], int bidx[8]) {
    const int half = lane >> 4;          // 0: K{0,1}, rows 0-7 of C;  1: K{2,3}, rows 8-15
    const int l16  = lane & 15;          // column / code-within-tile
#pragma unroll
    for (int r = 0; r < 8; ++r) { best[r] = 3.402823e38f; bidx[r] = 0; }

    const int ntiles = ncodes >> 4;
    for (int ct = 0; ct < ntiles; ++ct) {
        const float* row = cb + (size_t)(ct * 16 + l16) * RQ_C + 2 * half;
        // prefetch next code tile (same lane pattern), -> global_prefetch_b8
        __builtin_prefetch(row + 16 * RQ_C, 0, 0);

        v8f acc = {};                    // dot[m][n] accumulator, 16x16 f32
        float cp = 0.f;                  // partial ||c_n||^2 over this lane's K slots
#pragma unroll
        for (int s = 0; s < 16; ++s) {   // K = 64 in steps of 4
            v2f b = *(const v2f*)(row + 4 * s);
            cp = fmaf(b.x, b.x, fmaf(b.y, b.y, cp));
            // (neg_a, A, neg_b, B, c_mod, C, reuse_a, reuse_b)
            acc = __builtin_amdgcn_wmma_f32_16x16x4_f32(
                false, a[s], false, b, (short)0, acc, false, false);
        }
        // complete ||c||^2: lane L covers k=={0,1} mod 4, lane L^16 covers {2,3}
        const float csq = cp + __shfl_xor(cp, 16, 32);
        const int code = ct * 16 + l16;
#pragma unroll
        for (int r = 0; r < 8; ++r) {
            // dist - ||x||^2  =  ||c||^2 - 2*dot   (||x||^2 constant per row)
            const float sc = fmaf(-2.f, acc[r], csq);
            if (sc < best[r]) { best[r] = sc; bidx[r] = code; }
        }
    }
    // min-reduce across the 16 lanes of each half (xor<16 stays in-half)
#pragma unroll
    for (int off = 1; off < 16; off <<= 1) {
#pragma unroll
        for (int r = 0; r < 8; ++r) {
            const float ob = __shfl_xor(best[r], off, 32);
            const int   oi = __shfl_xor(bidx[r], off, 32);
            if (ob < best[r] || (ob == best[r] && oi < bidx[r])) {
                best[r] = ob; bidx[r] = oi;
            }
        }
    }
}

__global__ __launch_bounds__(32)
void ResidualQuantize_65000035057700_kernel(const float* __restrict__ X,
                                            const float* __restrict__ CB,
                                            const float* __restrict__ RCB,
                                            float* __restrict__ outQ,
                                            float* __restrict__ outIdx,
                                            float* __restrict__ outRQ,
                                            float* __restrict__ outIdxR) {
    __shared__ float s_res[16 * RQ_C];   // residual tile, row-major
    __shared__ int   s_idx[16];

    const int lane = threadIdx.x;
    const int half = lane >> 4;
    const int l16  = lane & 15;
    const int m0   = blockIdx.x * 16;

    // ---- Stage 1: load x tile into A-operand registers ----
    v2f a[16];
    {
        const float* xrow = X + (size_t)(m0 + l16) * RQ_C + 2 * half;
#pragma unroll
        for (int s = 0; s < 16; ++s) a[s] = *(const v2f*)(xrow + 4 * s);
    }

    float best[8]; int bidx[8];
    rq_scan(a, CB, RQ_K, lane, best, bidx);
    if (l16 == 0) {
#pragma unroll
        for (int r = 0; r < 8; ++r) s_idx[r + 8 * half] = bidx[r];
    }
    __syncthreads();

    // gather quantized rows, emit them, and build residual tile in LDS
#pragma unroll 4
    for (int m = 0; m < 16; ++m) {
        const int qi = s_idx[m];
        v2f q  = *(const v2f*)(CB + (size_t)qi * RQ_C + 2 * lane);
        v2f xv = *(const v2f*)(X + (size_t)(m0 + m) * RQ_C + 2 * lane);
        v2f r; r.x = xv.x - q.x; r.y = xv.y - q.y;
        *(v2f*)(outQ + (size_t)(m0 + m) * RQ_C + 2 * lane) = q;
        *(v2f*)(&s_res[m * RQ_C + 2 * lane]) = r;
    }
    if (lane < 16) outIdx[m0 + lane] = (float)s_idx[lane];
    __syncthreads();

    // ---- Stage 2: residual tile -> A layout, scan residual codebook ----
#pragma unroll
    for (int s = 0; s < 16; ++s)
        a[s] = *(const v2f*)(&s_res[l16 * RQ_C + 4 * s + 2 * half]);

    rq_scan(a, RCB, RQ_K, lane, best, bidx);
    if (l16 == 0) {
#pragma unroll
        for (int r = 0; r < 8; ++r) s_idx[r + 8 * half] = bidx[r];
    }
    __syncthreads();

#pragma unroll 4
    for (int m = 0; m < 16; ++m) {
        const int qi = s_idx[m];
        v2f q = *(const v2f*)(RCB + (size_t)qi * RQ_C + 2 * lane);
        *(v2f*)(outRQ + (size_t)(m0 + m) * RQ_C + 2 * lane) = q;
    }
    if (lane < 16) outIdxR[m0 + lane] = (float)s_idx[lane];
}

extern "C" void kernel_launch(void* const* d_in, const int* in_sizes, int n_in,
                              void* d_out, int out_size, void* d_ws, size_t ws_size,
                              hipStream_t stream) {
    (void)in_sizes; (void)n_in; (void)out_size; (void)d_ws; (void)ws_size;
    const float* X   = (const float*)d_in[0];
    const float* CB  = (const float*)d_in[1];
    const float* RCB = (const float*)d_in[2];

    float* outQ    = (float*)d_out;                    // [N, C]
    float* outIdx  = outQ  + (size_t)RQ_N * RQ_C;      // [N]
    float* outRQ   = outIdx + RQ_N;                    // [N, C]
    float* outIdxR = outRQ + (size_t)RQ_N * RQ_C;      // [N]

    ResidualQuantize_65000035057700_kernel<<<RQ_N / 16, 32, 0, stream>>>(
        X, CB, RCB, outQ, outIdx, outRQ, outIdxR);
}